// MaxPoolAggregation_29068338660043
// MI455X (gfx1250) — compile-verified
//
#include <hip/hip_runtime.h>
#include <stdint.h>

// Problem constants (from the reference)
#define N_NODES 8192
#define F_FEAT  128

// Tiling / mapping
#define TILE_J          64                     // neighbor columns staged per LDS tile
#define GROUPS          (TILE_J / 32)          // 32-bit ballot groups per tile = 2
#define ROWS_PER_WAVE   4
#define WAVES_PER_BLOCK 8
#define THREADS         (WAVES_PER_BLOCK * 32) // 256 threads = 8 wave32
#define ROWS_PER_BLOCK  (WAVES_PER_BLOCK * ROWS_PER_WAVE) // 32
#define TILE_FLOATS     (TILE_J * F_FEAT)      // 8192 floats = 32 KB
#define CHUNKS_PER_THR  ((TILE_FLOATS / 4) / THREADS)     // 16B chunks each = 8

#define NEG_HUGE (-3.402823466e38f)

// 16-byte vector type matching the builtin's parameter type (int __vector(4))
typedef __attribute__((__vector_size__(16))) int v4i_t;

// ---- CDNA5 async global->LDS copy (ASYNCcnt path), with safe fallback ----
__device__ __forceinline__ void async_copy_b128(const void* gsrc, void* ldst) {
#if __has_builtin(__builtin_amdgcn_global_load_async_to_lds_b128)
  __builtin_amdgcn_global_load_async_to_lds_b128(
      (__attribute__((address_space(1))) v4i_t*)gsrc,
      (__attribute__((address_space(3))) v4i_t*)ldst,
      /*offset=*/0, /*cpol=*/0);
#else
  *(float4*)ldst = *(const float4*)gsrc;  // synchronous fallback
#endif
}

__device__ __forceinline__ void wait_async_all() {
#if __has_builtin(__builtin_amdgcn_s_wait_asynccnt)
  __builtin_amdgcn_s_wait_asynccnt(0);
#else
  asm volatile("s_wait_asynccnt 0" ::: "memory");
#endif
}

__device__ __forceinline__ uint32_t ballot32(bool p) {
#if __has_builtin(__builtin_amdgcn_ballot_w32)
  return __builtin_amdgcn_ballot_w32(p);
#else
  return (uint32_t)__ballot(p);
#endif
}

__global__ __launch_bounds__(THREADS)
void MaxPoolAggregation_kernel(const float* __restrict__ X,
                               const int*   __restrict__ adj,
                               float*       __restrict__ out) {
  // Double-buffered X tile: 2 * 32 KB = 64 KB LDS
  __shared__ float lds[2 * TILE_FLOATS];

  const int tid     = (int)threadIdx.x;
  const int lane    = tid & 31;       // lane -> feature group (4 floats each)
  const int wave    = tid >> 5;       // wave -> row group
  const int rowBase = (int)blockIdx.x * ROWS_PER_BLOCK + wave * ROWS_PER_WAVE;

  float* buf0 = lds;
  float* buf1 = lds + TILE_FLOATS;

  float4 acc[ROWS_PER_WAVE];
  uint32_t anyN[ROWS_PER_WAVE];
#pragma unroll
  for (int r = 0; r < ROWS_PER_WAVE; ++r) {
    acc[r].x = NEG_HUGE; acc[r].y = NEG_HUGE;
    acc[r].z = NEG_HUGE; acc[r].w = NEG_HUGE;
    anyN[r] = 0u;
  }

  // Prefetch tile 0 into buf0 (async, all 256 threads cooperate, coalesced 16B chunks)
  {
    const float* src = X;  // tile 0 starts at X[0]
#pragma unroll
    for (int k = 0; k < CHUNKS_PER_THR; ++k) {
      const int c = tid + k * THREADS;          // 16-byte chunk index
      async_copy_b128(src + c * 4, buf0 + c * 4);
    }
  }

  const int nTiles = N_NODES / TILE_J;          // 128

#pragma unroll 1
  for (int t = 0; t < nTiles; ++t) {
    float* cur = (t & 1) ? buf1 : buf0;
    float* nxt = (t & 1) ? buf0 : buf1;

    wait_async_all();      // this wave's async loads into 'cur' complete
    __syncthreads();       // all waves' loads visible; prior compute on 'nxt' done

    // Kick off async prefetch of the next tile while we compute on 'cur'
    if (t + 1 < nTiles) {
      const float* src = X + (size_t)(t + 1) * TILE_FLOATS;
#pragma unroll
      for (int k = 0; k < CHUNKS_PER_THR; ++k) {
        const int c = tid + k * THREADS;
        async_copy_b128(src + c * 4, nxt + c * 4);
      }
    }

    // Build per-row neighbor bitmasks for this tile (adj is the 256 MB stream:
    // non-temporal so it does not evict X from L2). Bit l of m[r][g] corresponds
    // to neighbor j = j0 + g*32 + l, uniform across the wave -> SGPR masks.
    const int j0 = t * TILE_J;
    uint32_t m[ROWS_PER_WAVE][GROUPS];
#pragma unroll
    for (int r = 0; r < ROWS_PER_WAVE; ++r) {
      const int* arow = adj + (size_t)(rowBase + r) * N_NODES + j0;
#pragma unroll
      for (int g = 0; g < GROUPS; ++g) {
        const int a = __builtin_nontemporal_load(arow + g * 32 + lane);
        m[r][g] = ballot32(a != 0);
        anyN[r] |= m[r][g];
      }
    }

    // Masked max: one ds_load_b128 of X[j] per j, reused by 4 row accumulators.
    // The bit test is uniform -> scalar branch (SALU pipe) skips ~50% of maxes.
#pragma unroll 1
    for (int g = 0; g < GROUPS; ++g) {
#pragma unroll 4
      for (int b = 0; b < 32; ++b) {
        const float4 x =
            *(const float4*)(cur + (g * 32 + b) * F_FEAT + lane * 4);
        const uint32_t bit = 1u << b;
#pragma unroll
        for (int r = 0; r < ROWS_PER_WAVE; ++r) {
          if (m[r][g] & bit) {
            acc[r].x = fmaxf(acc[r].x, x.x);
            acc[r].y = fmaxf(acc[r].y, x.y);
            acc[r].z = fmaxf(acc[r].z, x.z);
            acc[r].w = fmaxf(acc[r].w, x.w);
          }
        }
      }
    }
  }

  // Emit: rows with no neighbors produce zeros (reference semantics).
#pragma unroll
  for (int r = 0; r < ROWS_PER_WAVE; ++r) {
    float4 v;
    if (anyN[r]) {
      v = acc[r];
    } else {
      v.x = 0.0f; v.y = 0.0f; v.z = 0.0f; v.w = 0.0f;
    }
    *(float4*)(out + (size_t)(rowBase + r) * F_FEAT + lane * 4) = v;
  }
}

extern "C" void kernel_launch(void* const* d_in, const int* in_sizes, int n_in,
                              void* d_out, int out_size, void* d_ws, size_t ws_size,
                              hipStream_t stream) {
  (void)in_sizes; (void)n_in; (void)d_ws; (void)ws_size; (void)out_size;
  const float* X   = (const float*)d_in[0];   // (8192, 128) f32
  const int*   adj = (const int*)d_in[1];     // (8192, 8192) i32
  float*       out = (float*)d_out;           // (8192, 128) f32

  dim3 grid(N_NODES / ROWS_PER_BLOCK);        // 256 blocks
  dim3 block(THREADS);                        // 256 threads = 8 wave32
  MaxPoolAggregation_kernel<<<grid, block, 0, stream>>>(X, adj, out);
}